// TightlyPackedTaylorTerm_3599182594293
// MI455X (gfx1250) — compile-verified
//
#include <hip/hip_runtime.h>
#include <hip/hip_bf16.h>

// Causal degree-2 polynomial attention ("tightly packed Taylor term", P=2).
// Identity: sum_m C[m] * phi_m(q) * phi_m(k) = (q . k)^2 -> never materialize
// the 528-dim monomial features. M, C, continue_prev are unused.
//
//   Y_S[h,s,:] = ALPHA * sum_{t<=s} (q_s . k_t)^2 * v_t      [H,S,DV]
//   Y_Z[h,s]   = ALPHA * sum_{t<=s} (q_s . k_t)^2            [H,S,1]
//
// fp32 inputs -> V_WMMA_F32_16X16X4_F32 (full-precision matrix op).
// Y_Z is computed as P @ ones via a 5th WMMA accumulator with a constant
// all-ones B fragment (no cross-lane shuffles needed).

#define NH     8
#define SEQ    512
#define DK     32
#define DV     64
#define BM     128          // query rows per workgroup (8 waves x 16)
#define BN     16           // key rows per inner block
#define ALPHA  (1.0f / 64.0f)  // 1/(2! * 32^(2/2))

typedef float v2f __attribute__((ext_vector_type(2)));
typedef float v8f __attribute__((ext_vector_type(8)));

__device__ __forceinline__ v8f wmma_f32_k4(v2f a, v2f b, v8f c) {
    // 8 args: (neg_a, A, neg_b, B, c_mod, C, reuse_a, reuse_b)
    return __builtin_amdgcn_wmma_f32_16x16x4_f32(false, a, false, b,
                                                 (short)0, c, false, false);
}

__global__ __launch_bounds__(256)
void poly2_attn_kernel(const float* __restrict__ Q,
                       const float* __restrict__ K,
                       const float* __restrict__ V,
                       float* __restrict__ YS,
                       float* __restrict__ YZ) {
    // +1 padding per row: odd dword stride -> conflict-free strided reads
    __shared__ float Qs[BM][DK + 1];
    __shared__ float Ks[BN][DK + 1];
    __shared__ float Vs[BN][DV + 1];
    __shared__ float Ps[8][16][17];   // per-wave P tile (C-layout -> A-layout)

    const int h    = blockIdx.y;
    const int q0   = blockIdx.x * BM;
    const int tid  = threadIdx.x;
    const int wave = tid >> 5;
    const int lane = tid & 31;
    const int r    = lane & 15;      // row/col-within-half index
    const int hi   = lane >> 4;      // which 16-lane half

    const float* Qh = Q + (size_t)h * SEQ * DK;
    const float* Kh = K + (size_t)h * SEQ * DK;
    const float* Vh = V + (size_t)h * SEQ * DV;

    // Cooperative load of this block's Q tile (128 x 32)
    for (int idx = tid; idx < BM * DK; idx += 256) {
        const int row = idx >> 5, col = idx & 31;
        Qs[row][col] = Qh[(size_t)(q0 + row) * DK + col];
    }
    __syncthreads();

    const int qrow_local = wave * 16;         // this wave's query tile
    const int qbase      = q0 + qrow_local;   // global first query row
    const int jd         = qbase >> 4;        // diagonal key-block index
    const int nj         = (q0 >> 4) + (BM >> 4);  // key blocks for this WG

    // Q A-fragments are invariant over the whole key loop: hold in registers.
    v2f qa[DK / 4];
#pragma unroll
    for (int kk = 0; kk < DK; kk += 4) {
        const int c0 = kk + 2 * hi;
        qa[kk >> 2] = (v2f){ Qs[qrow_local + r][c0], Qs[qrow_local + r][c0 + 1] };
    }

    v8f acc0 = {0,0,0,0,0,0,0,0};
    v8f acc1 = {0,0,0,0,0,0,0,0};
    v8f acc2 = {0,0,0,0,0,0,0,0};
    v8f acc3 = {0,0,0,0,0,0,0,0};
    v8f accz = {0,0,0,0,0,0,0,0};    // Y_Z = P @ ones (all columns identical)
    const v2f ones = {1.0f, 1.0f};

    for (int j = 0; j < nj; ++j) {
        const int k0 = j * BN;
        __syncthreads();   // protect Ks/Vs/Ps from readers of previous iter
        for (int idx = tid; idx < BN * DK; idx += 256) {
            const int row = idx >> 5, col = idx & 31;
            Ks[row][col] = Kh[(size_t)(k0 + row) * DK + col];
        }
        for (int idx = tid; idx < BN * DV; idx += 256) {
            const int row = idx >> 6, col = idx & 63;
            Vs[row][col] = Vh[(size_t)(k0 + row) * DV + col];
        }
        // Prefetch next key block into cache while this one is consumed.
        if (j + 1 < nj && tid < BN) {
            const int kn = k0 + BN + tid;
            __builtin_prefetch(&Kh[(size_t)kn * DK], 0, 1);
            __builtin_prefetch(&Vh[(size_t)kn * DV], 0, 1);
            __builtin_prefetch(&Vh[(size_t)kn * DV + 32], 0, 1);
        }
        __syncthreads();

        const bool active = (j <= jd);   // wave-uniform: EXEC stays all-ones
        if (active) {
            // ---- S = Q_tile . K_tile^T : batch all B-fragment loads,
            //      then chain 8 fp32 WMMAs (one dscnt wait total) ----
            v2f kb[DK / 4];
#pragma unroll
            for (int kk = 0; kk < DK; kk += 4) {
                const int c0 = kk + 2 * hi;
                kb[kk >> 2] = (v2f){ Ks[r][c0], Ks[r][c0 + 1] };
            }
            v8f s = {0,0,0,0,0,0,0,0};
#pragma unroll
            for (int t = 0; t < DK / 4; ++t)
                s = wmma_f32_k4(qa[t], kb[t], s);

            // ---- square, causal mask, stage P in LDS ----
#pragma unroll
            for (int i = 0; i < 8; ++i) {
                float p = s[i] * s[i];
                if (j == jd) {
                    const int gq = qbase + i + 8 * hi;   // C-layout row
                    const int gk = k0 + r;               // C-layout col
                    if (gk > gq) p = 0.0f;
                }
                Ps[wave][i + 8 * hi][r] = p;
            }
        }
        __syncthreads();   // P visible (and keeps waves in lockstep)

        if (active) {
            // ---- Y += P . V_tile ; Y_Z += P . ones ----
#pragma unroll
            for (int kk = 0; kk < BN; kk += 4) {
                const int c0 = kk + 2 * hi;
                v2f a  = { Ps[wave][r][c0], Ps[wave][r][c0 + 1] };
                v2f b0 = { Vs[c0][ 0 + r], Vs[c0 + 1][ 0 + r] };
                v2f b1 = { Vs[c0][16 + r], Vs[c0 + 1][16 + r] };
                v2f b2 = { Vs[c0][32 + r], Vs[c0 + 1][32 + r] };
                v2f b3 = { Vs[c0][48 + r], Vs[c0 + 1][48 + r] };
                acc0 = wmma_f32_k4(a, b0, acc0);
                acc1 = wmma_f32_k4(a, b1, acc1);
                acc2 = wmma_f32_k4(a, b2, acc2);
                acc3 = wmma_f32_k4(a, b3, acc3);
                accz = wmma_f32_k4(a, ones, accz);
            }
        }
    }

    // ---- writeback: Y_S [h, s, d], Y_Z [h, s] ----
    float* ys = YS + ((size_t)h * SEQ + qbase) * DV;
#pragma unroll
    for (int i = 0; i < 8; ++i) {
        const int row = i + 8 * hi;          // C-layout row of this lane
        ys[(size_t)row * DV +  0 + r] = acc0[i] * ALPHA;
        ys[(size_t)row * DV + 16 + r] = acc1[i] * ALPHA;
        ys[(size_t)row * DV + 32 + r] = acc2[i] * ALPHA;
        ys[(size_t)row * DV + 48 + r] = acc3[i] * ALPHA;
    }
    if (r == 0) {                             // lanes 0 and 16: rows 0-7 / 8-15
#pragma unroll
        for (int i = 0; i < 8; ++i) {
            YZ[(size_t)h * SEQ + qbase + i + 8 * hi] = accz[i] * ALPHA;
        }
    }
}

extern "C" void kernel_launch(void* const* d_in, const int* in_sizes, int n_in,
                              void* d_out, int out_size, void* d_ws, size_t ws_size,
                              hipStream_t stream) {
    (void)in_sizes; (void)n_in; (void)out_size; (void)d_ws; (void)ws_size;
    const float* Q = (const float*)d_in[0];
    const float* K = (const float*)d_in[1];
    const float* V = (const float*)d_in[2];
    // d_in[3] = M, d_in[4] = C, d_in[5] = continue_prev: unused (see header)
    float* YS = (float*)d_out;                              // [1,8,512,64]
    float* YZ = (float*)d_out + (size_t)NH * SEQ * DV;      // [1,8,512,1]

    dim3 grid(SEQ / BM, NH);
    dim3 block(256);
    hipLaunchKernelGGL(poly2_attn_kernel, grid, block, 0, stream, Q, K, V, YS, YZ);
}